// MoELayer_79706003079905
// MI455X (gfx1250) — compile-verified
//
#include <hip/hip_runtime.h>
#include <math.h>
#include <stdint.h>

typedef float v2f __attribute__((ext_vector_type(2)));
typedef float v8f __attribute__((ext_vector_type(8)));

#define N_TOK 8192
#define DDIM  1024
#define HDIM  1024
#define NEXP  8
#define CAPF  1280.0f   // max(int(8192*1.25/8), 4)

#if defined(__gfx1250__) && \
    __has_builtin(__builtin_amdgcn_global_load_async_to_lds_b128) && \
    __has_builtin(__builtin_amdgcn_s_wait_asynccnt)
#define USE_ASYNC 1
#else
#define USE_ASYNC 0
#endif

#if USE_ASYNC
typedef int b128_t __attribute__((vector_size(16)));
typedef __attribute__((address_space(1))) b128_t gb128_t;   // global (as1)
typedef __attribute__((address_space(3))) b128_t lb128_t;   // LDS (as3)
#define GPTR(p) ((gb128_t*)(uintptr_t)(p))
#define SPTR(p) ((lb128_t*)(uintptr_t)(p))
#endif

// ----------------------------------------------------------------------------
// zero output
// ----------------------------------------------------------------------------
__global__ void zero_f32(float* __restrict__ p, int n) {
  int i = blockIdx.x * blockDim.x + threadIdx.x;
  if (i < n) p[i] = 0.0f;
}

// ----------------------------------------------------------------------------
// router: logits -> softmax -> top2 -> normalized probs -> mask "set" rows
// ----------------------------------------------------------------------------
__global__ __launch_bounds__(256) void router_kernel(
    const float* __restrict__ x,    // N x D
    const float* __restrict__ rw,   // D x E
    const float* __restrict__ rb,   // E
    float* __restrict__ mask,       // N x E
    int*   __restrict__ topi,       // N x 2
    float* __restrict__ topp) {     // N x 2
  __shared__ float xs[DDIM];
  __shared__ float lg[NEXP];
  const int n = blockIdx.x;
  const int tid = threadIdx.x;
  #pragma unroll
  for (int i = 0; i < DDIM / 256; ++i)
    xs[tid + i * 256] = x[(size_t)n * DDIM + tid + i * 256];
  __syncthreads();
  const int w = tid >> 5, lane = tid & 31;
  float s = 0.f;
  for (int d = lane; d < DDIM; d += 32) s += xs[d] * rw[d * NEXP + w];
  #pragma unroll
  for (int off = 16; off > 0; off >>= 1) s += __shfl_xor(s, off, 32);
  if (lane == 0) lg[w] = s + rb[w];
  __syncthreads();
  if (tid == 0) {
    float mx = lg[0];
    #pragma unroll
    for (int e = 1; e < NEXP; ++e) mx = fmaxf(mx, lg[e]);
    float p[NEXP], sum = 0.f;
    #pragma unroll
    for (int e = 0; e < NEXP; ++e) { p[e] = expf(lg[e] - mx); sum += p[e]; }
    const float inv = 1.0f / sum;
    #pragma unroll
    for (int e = 0; e < NEXP; ++e) p[e] *= inv;
    int i0 = 0;
    #pragma unroll
    for (int e = 1; e < NEXP; ++e) if (p[e] > p[i0]) i0 = e;   // ties -> lowest idx
    int i1 = (i0 == 0) ? 1 : 0;
    #pragma unroll
    for (int e = 0; e < NEXP; ++e) if (e != i0 && p[e] > p[i1]) i1 = e;
    const float s2 = p[i0] + p[i1];
    const float p0 = p[i0] / s2, p1 = p[i1] / s2;
    #pragma unroll
    for (int e = 0; e < NEXP; ++e) mask[(size_t)n * NEXP + e] = 0.f;
    mask[(size_t)n * NEXP + i0] = p0;
    mask[(size_t)n * NEXP + i1] = p1;
    topi[n * 2 + 0] = i0; topi[n * 2 + 1] = i1;
    topp[n * 2 + 0] = p0; topp[n * 2 + 1] = p1;
  }
}

// replicate torch scatter_add_(0, top_i, top_p): mask[top_i[n,j], j] += top_p[n,j]
__global__ void scatter_bug_kernel(float* __restrict__ mask,
                                   const int* __restrict__ topi,
                                   const float* __restrict__ topp) {
  int n = blockIdx.x * blockDim.x + threadIdx.x;
  if (n < N_TOK) {
    #pragma unroll
    for (int j = 0; j < 2; ++j)
      atomicAdd(&mask[(size_t)topi[n * 2 + j] * NEXP + j], topp[n * 2 + j]);
  }
}

__global__ void clamp_kernel(float* __restrict__ mask) {
  int i = blockIdx.x * blockDim.x + threadIdx.x;
  if (i < N_TOK * NEXP) mask[i] = fminf(mask[i], CAPF);
}

// ----------------------------------------------------------------------------
// FFN GEMM: C(8192x1024) = A(8192x1024) @ W(1024x1024), fp32 WMMA 16x16x4.
// Block tile 128x128, BK=16, double-buffered LDS fed by async global->LDS.
// 8 waves in 2x4 grid; wave tile 64x32 (4x2 accumulators).
// EPI==1: C = mask * gelu(mask*acc + b1)      (h scratch, combine weight folded)
// EPI==2: C += acc + mask * b2                (accumulate into final output)
// ----------------------------------------------------------------------------
template <int EPI>
__global__ __launch_bounds__(256) void ffn_gemm(
    const float* __restrict__ A,     // N x 1024
    const float* __restrict__ W,     // 1024 x 1024
    const float* __restrict__ bias,  // 1024 (per output col)
    const float* __restrict__ mask,  // N x E
    float* __restrict__ C,           // N x 1024
    int e) {
#if USE_ASYNC
  __shared__ float As[2][128][20];   // row-major M x K, row padded to 80B
  __shared__ float Bs[2][16][132];   // K x N, row padded to 528B
#else
  __shared__ float As[1][128][20];
  __shared__ float Bs[1][16][132];
#endif
  __shared__ float maskS[128];
  __shared__ float biasS[128];

  const int tid = threadIdx.x;
  const int m0 = blockIdx.y * 128;
  const int n0 = blockIdx.x * 128;

  if (tid < 128) {
    maskS[tid] = mask[(size_t)(m0 + tid) * NEXP + e];
    biasS[tid] = bias[n0 + tid];
  }

  const int wid  = tid >> 5;
  const int lane = tid & 31;
  const int lhi  = lane >> 4;        // half-wave select (K pair / M+8)
  const int lm   = lane & 15;
  const int wm   = (wid >> 2) * 64;  // wave row origin in block tile
  const int wn   = (wid & 3) * 32;   // wave col origin in block tile

  v8f acc[4][2];
  #pragma unroll
  for (int i = 0; i < 4; ++i)
    #pragma unroll
    for (int j = 0; j < 2; ++j)
      #pragma unroll
      for (int r = 0; r < 8; ++r) acc[i][j][r] = 0.0f;

  // 4 WMMA k-steps (K=4 each) on one resident tile pair
  auto computeTile = [&](const float (&Asb)[128][20], const float (&Bsb)[16][132]) {
    #pragma unroll
    for (int ks = 0; ks < 4; ++ks) {
      const int kb = ks * 4 + 2 * lhi;  // lanes 0-15: K {kb,kb+1}; 16-31: {kb+2,kb+3}
      v2f a[4], b[2];
      #pragma unroll
      for (int i = 0; i < 4; ++i)       // single aligned ds_load_b64 per fragment
        a[i] = *reinterpret_cast<const v2f*>(&Asb[wm + i * 16 + lm][kb]);
      #pragma unroll
      for (int j = 0; j < 2; ++j) {
        b[j][0] = Bsb[kb + 0][wn + j * 16 + lm];
        b[j][1] = Bsb[kb + 1][wn + j * 16 + lm];
      }
      #pragma unroll
      for (int i = 0; i < 4; ++i)
        #pragma unroll
        for (int j = 0; j < 2; ++j)
          acc[i][j] = __builtin_amdgcn_wmma_f32_16x16x4_f32(
              false, a[i], false, b[j], (short)0, acc[i][j], false, false);
    }
  };

#if USE_ASYNC
  // each thread issues 4 async b128 copies per tile (2 for A, 2 for B)
  auto issueTile = [&](int kc, int buf) {
    const int k0 = kc * 16;
    #pragma unroll
    for (int i = 0; i < 2; ++i) {
      const int idx = tid + i * 256;       // 512 x 16B chunks of the A tile
      const int r   = idx >> 2;            // 0..127
      const int cb  = (idx & 3) * 16;      // byte offset within 64B row
      __builtin_amdgcn_global_load_async_to_lds_b128(
          GPTR((const char*)(A + (size_t)(m0 + r) * DDIM + k0) + cb),
          SPTR((char*)&As[buf][r][0] + cb), 0, 0);
    }
    #pragma unroll
    for (int i = 0; i < 2; ++i) {
      const int idx = tid + i * 256;       // 512 x 16B chunks of the B tile
      const int kr  = idx >> 5;            // 0..15
      const int cb  = (idx & 31) * 16;     // byte offset within 512B row
      __builtin_amdgcn_global_load_async_to_lds_b128(
          GPTR((const char*)(W + (size_t)(k0 + kr) * HDIM + n0) + cb),
          SPTR((char*)&Bs[buf][kr][0] + cb), 0, 0);
    }
  };

  issueTile(0, 0);
  for (int kc = 0; kc < DDIM / 16; ++kc) {
    const int buf = kc & 1;
    if (kc + 1 < DDIM / 16) {
      issueTile(kc + 1, buf ^ 1);               // prefetch next tile
      __builtin_amdgcn_s_wait_asynccnt(4);      // current tile's 4 copies done
    } else {
      __builtin_amdgcn_s_wait_asynccnt(0);
    }
    __syncthreads();
    computeTile(As[buf], Bs[buf]);
    __syncthreads();                            // buffer reusable for tile kc+2
  }
#else
  for (int kc = 0; kc < DDIM / 16; ++kc) {
    const int k0 = kc * 16;
    #pragma unroll
    for (int i = 0; i < 2; ++i) {
      const int idx = tid + i * 256;
      const int r   = idx >> 2;
      const int c4  = (idx & 3) * 4;
      const float4 v =
          *reinterpret_cast<const float4*>(&A[(size_t)(m0 + r) * DDIM + k0 + c4]);
      *reinterpret_cast<float4*>(&As[0][r][c4]) = v;
      if (kc + 1 < DDIM / 16)
        __builtin_prefetch(&A[(size_t)(m0 + r) * DDIM + k0 + 16 + c4], 0, 1);
    }
    #pragma unroll
    for (int i = 0; i < 2; ++i) {
      const int idx = tid + i * 256;
      const int kr  = idx >> 5;
      const int cc  = (idx & 31) * 4;
      const float4 v =
          *reinterpret_cast<const float4*>(&W[(size_t)(k0 + kr) * HDIM + n0 + cc]);
      *reinterpret_cast<float4*>(&Bs[0][kr][cc]) = v;
      if (kc + 1 < DDIM / 16)
        __builtin_prefetch(&W[(size_t)(k0 + 16 + kr) * HDIM + n0 + cc], 0, 1);
    }
    __syncthreads();
    computeTile(As[0], Bs[0]);
    __syncthreads();
  }
#endif

  // ---- epilogue ----
  #pragma unroll
  for (int i = 0; i < 4; ++i) {
    #pragma unroll
    for (int j = 0; j < 2; ++j) {
      #pragma unroll
      for (int r = 0; r < 8; ++r) {
        const int row = wm + i * 16 + r + 8 * lhi;  // C layout: M = r + 8*(lane>=16)
        const int col = wn + j * 16 + lm;
        const size_t g = (size_t)(m0 + row) * HDIM + (n0 + col);
        const float v = acc[i][j][r];
        if (EPI == 1) {
          const float mk  = maskS[row];
          const float pre = mk * v + biasS[col];
          const float gel = 0.5f * pre * (1.0f + erff(pre * 0.70710678118654752f));
          C[g] = mk * gel;                       // fold combine weight into h
        } else {
          C[g] += v + maskS[row] * biasS[col];   // += h@w2 + mask*b2
        }
      }
    }
  }
}

// ----------------------------------------------------------------------------
extern "C" void kernel_launch(void* const* d_in, const int* in_sizes, int n_in,
                              void* d_out, int out_size, void* d_ws, size_t ws_size,
                              hipStream_t stream) {
  const float* x  = (const float*)d_in[0];  // (B,S,D) -> (N,D)
  const float* rw = (const float*)d_in[1];  // (D,E)
  const float* rb = (const float*)d_in[2];  // (E)
  const float* w1 = (const float*)d_in[3];  // (E,D,H)
  const float* b1 = (const float*)d_in[4];  // (E,H)
  const float* w2 = (const float*)d_in[5];  // (E,H,D)
  const float* b2 = (const float*)d_in[6];  // (E,D)
  float* out = (float*)d_out;

  char* ws = (char*)d_ws;
  float* mask = (float*)ws;                          // 256 KB
  int*   topi = (int*)(ws + 262144);                 // 64 KB
  float* topp = (float*)(ws + 262144 + 65536);       // 64 KB
  float* Htmp = (float*)(ws + 524288);               // 32 MB (N x H)

  zero_f32<<<(N_TOK * DDIM) / 256, 256, 0, stream>>>(out, N_TOK * DDIM);
  router_kernel<<<N_TOK, 256, 0, stream>>>(x, rw, rb, mask, topi, topp);
  scatter_bug_kernel<<<N_TOK / 256, 256, 0, stream>>>(mask, topi, topp);
  clamp_kernel<<<(N_TOK * NEXP) / 256, 256, 0, stream>>>(mask);

  dim3 grid(HDIM / 128, N_TOK / 128);  // (8, 64)
  for (int e = 0; e < NEXP; ++e) {
    ffn_gemm<1><<<grid, 256, 0, stream>>>(
        x, w1 + (size_t)e * DDIM * HDIM, b1 + (size_t)e * HDIM, mask, Htmp, e);
    ffn_gemm<2><<<grid, 256, 0, stream>>>(
        Htmp, w2 + (size_t)e * HDIM * DDIM, b2 + (size_t)e * DDIM, mask, out, e);
  }
}